// RGCN_11038065950752
// MI455X (gfx1250) — compile-verified
//
#include <hip/hip_runtime.h>
#include <math.h>

// ---------------- problem constants (match reference) ----------------
enum : int {
  N_LOANS   = 300000,
  N_CLIENTS = 150000,
  FD        = 128,      // feature dim
  E_CL      = 600000,
  E_LC      = 600000,
  E_LL      = 400000,
};

#define LDSTRIDE 132   // 128 + 4 pad -> conflict-free fragment reads

typedef __attribute__((ext_vector_type(2))) float v2f;
typedef __attribute__((ext_vector_type(8))) float v8f;

__device__ __forceinline__ void fatomic_add(float* p, float v) {
  (void)unsafeAtomicAdd(p, v);   // hardware global_atomic_add_f32
}

// ---------------- degree histogram ----------------
__global__ void count_deg_kernel(const int* __restrict__ idx, int n,
                                 float* __restrict__ deg) {
  int i = blockIdx.x * blockDim.x + threadIdx.x;
  if (i < n) fatomic_add(&deg[idx[i]], 1.0f);
}

// deg -> rsqrt(max(deg,1)) in place (run over all 6 arrays at once)
__global__ void rsqrt_clip_kernel(float* __restrict__ deg, int n) {
  int i = blockIdx.x * blockDim.x + threadIdx.x;
  if (i < n) {
    float d = deg[i];
    deg[i] = rsqrtf(d < 1.0f ? 1.0f : d);
  }
}

// ---------------- SpMM: agg[dst] += x[src] * rs_src[src] ----------------
// one wave (32 lanes) per edge, 4 floats per lane (float4 gather)
__global__ void spmm_kernel(const float* __restrict__ x,
                            const float* __restrict__ rs_src,
                            const int* __restrict__ src,
                            const int* __restrict__ dst,
                            int nedges,
                            float* __restrict__ agg) {
  int e = blockIdx.x * 8 + (threadIdx.x >> 5);
  if (e >= nedges) return;
  int lane = threadIdx.x & 31;
  int s = src[e];
  int d = dst[e];
  float scale = rs_src[s];
  float4 v = *((const float4*)(x + (size_t)s * FD) + lane);
  float* ap = agg + (size_t)d * FD + (size_t)lane * 4;
  fatomic_add(ap + 0, v.x * scale);
  fatomic_add(ap + 1, v.y * scale);
  fatomic_add(ap + 2, v.z * scale);
  fatomic_add(ap + 3, v.w * scale);
}

// ---------------- WMMA GEMM, clients pass ----------------
// one block = 16 rows, 8 waves cover the 128 output columns.
// h_out = relu((rs*agg) @ W1 + b1) ; o_out = (rs*agg) @ W2 + b2
__global__ void gemm_clients_kernel(const float* __restrict__ agg,
                                    const float* __restrict__ rs_in,
                                    const float* __restrict__ W1,
                                    const float* __restrict__ b1,
                                    const float* __restrict__ W2,
                                    const float* __restrict__ b2,
                                    float* __restrict__ h_out,
                                    float* __restrict__ o_out) {
  __shared__ float tA[16 * LDSTRIDE];
  const int tid  = threadIdx.x;
  const int row0 = blockIdx.x * 16;

  #pragma unroll
  for (int j = 0; j < 8; ++j) {
    int i = tid + j * 256;
    int r = i >> 7;
    int k = i & 127;
    tA[r * LDSTRIDE + k] = agg[(size_t)(row0 + r) * FD + k] * rs_in[row0 + r];
  }
  __syncthreads();

  const int lane = tid & 31;
  const int wave = tid >> 5;
  const int m    = lane & 15;
  const int hi   = lane >> 4;           // 0 or 1
  const int n    = wave * 16 + m;       // output column

  v8f c1 = {};
  v8f c2 = {};
  #pragma unroll 4
  for (int kk = 0; kk < FD; kk += 4) {
    int kb = kk + hi * 2;
    v2f a;
    a.x = tA[m * LDSTRIDE + kb];
    a.y = tA[m * LDSTRIDE + kb + 1];
    v2f bb1;
    bb1.x = W1[kb * FD + n];
    bb1.y = W1[(kb + 1) * FD + n];
    v2f bb2;
    bb2.x = W2[kb * FD + n];
    bb2.y = W2[(kb + 1) * FD + n];
    c1 = __builtin_amdgcn_wmma_f32_16x16x4_f32(false, a, false, bb1, (short)0, c1, false, false);
    c2 = __builtin_amdgcn_wmma_f32_16x16x4_f32(false, a, false, bb2, (short)0, c2, false, false);
  }

  float bias1 = b1[n];
  float bias2 = b2[n];
  #pragma unroll
  for (int v = 0; v < 8; ++v) {
    int row = row0 + v + hi * 8;
    float h = c1[v] + bias1;
    h_out[(size_t)row * FD + n] = h > 0.0f ? h : 0.0f;
    o_out[(size_t)row * FD + n] = c2[v] + bias2;
  }
}

// ---------------- WMMA GEMM, loans pass (two A tiles summed) ----------------
// out = (rsA*aggA) @ WA + (rsB*aggB) @ WB + bA + bB
__global__ void gemm_loans_kernel(const float* __restrict__ aggA,
                                  const float* __restrict__ rsA,
                                  const float* __restrict__ WA,
                                  const float* __restrict__ aggB,
                                  const float* __restrict__ rsB,
                                  const float* __restrict__ WB,
                                  const float* __restrict__ bA,
                                  const float* __restrict__ bB,
                                  float* __restrict__ out) {
  __shared__ float tA[16 * LDSTRIDE];
  __shared__ float tB[16 * LDSTRIDE];
  const int tid  = threadIdx.x;
  const int row0 = blockIdx.x * 16;

  #pragma unroll
  for (int j = 0; j < 8; ++j) {
    int i = tid + j * 256;
    int r = i >> 7;
    int k = i & 127;
    tA[r * LDSTRIDE + k] = aggA[(size_t)(row0 + r) * FD + k] * rsA[row0 + r];
    tB[r * LDSTRIDE + k] = aggB[(size_t)(row0 + r) * FD + k] * rsB[row0 + r];
  }
  __syncthreads();

  const int lane = tid & 31;
  const int wave = tid >> 5;
  const int m    = lane & 15;
  const int hi   = lane >> 4;
  const int n    = wave * 16 + m;

  v8f c = {};
  #pragma unroll 4
  for (int kk = 0; kk < FD; kk += 4) {
    int kb = kk + hi * 2;
    v2f aA;
    aA.x = tA[m * LDSTRIDE + kb];
    aA.y = tA[m * LDSTRIDE + kb + 1];
    v2f bA_;
    bA_.x = WA[kb * FD + n];
    bA_.y = WA[(kb + 1) * FD + n];
    c = __builtin_amdgcn_wmma_f32_16x16x4_f32(false, aA, false, bA_, (short)0, c, false, false);

    v2f aB;
    aB.x = tB[m * LDSTRIDE + kb];
    aB.y = tB[m * LDSTRIDE + kb + 1];
    v2f bB_;
    bB_.x = WB[kb * FD + n];
    bB_.y = WB[(kb + 1) * FD + n];
    c = __builtin_amdgcn_wmma_f32_16x16x4_f32(false, aB, false, bB_, (short)0, c, false, false);
  }

  float bias = bA[n] + bB[n];
  #pragma unroll
  for (int v = 0; v < 8; ++v) {
    int row = row0 + v + hi * 8;
    out[(size_t)row * FD + n] = c[v] + bias;
  }
}

// ---------------- launcher ----------------
extern "C" void kernel_launch(void* const* d_in, const int* in_sizes, int n_in,
                              void* d_out, int out_size, void* d_ws, size_t ws_size,
                              hipStream_t stream) {
  const float* x_loans = (const float*)d_in[0];
  const int* src_cl = (const int*)d_in[2];
  const int* dst_cl = (const int*)d_in[3];
  const int* src_lc = (const int*)d_in[4];
  const int* dst_lc = (const int*)d_in[5];
  const int* src_ll = (const int*)d_in[6];
  const int* dst_ll = (const int*)d_in[7];
  const float* W1_lc = (const float*)d_in[9];
  const float* W2_cl = (const float*)d_in[11];
  const float* W2_lc = (const float*)d_in[12];
  const float* W2_ll = (const float*)d_in[13];
  const float* b1_lc = (const float*)d_in[15];
  const float* b2_cl = (const float*)d_in[17];
  const float* b2_lc = (const float*)d_in[18];
  const float* b2_ll = (const float*)d_in[19];

  // workspace layout (floats); zeroed region first
  float* ws = (float*)d_ws;
  float* rs_out_lc = ws;                          // N_LOANS
  float* rs_in_lc  = rs_out_lc + N_LOANS;         // N_CLIENTS
  float* rs_out_cl = rs_in_lc  + N_CLIENTS;       // N_CLIENTS
  float* rs_in_cl  = rs_out_cl + N_CLIENTS;       // N_LOANS
  float* rs_out_ll = rs_in_cl  + N_LOANS;         // N_LOANS
  float* rs_in_ll  = rs_out_ll + N_LOANS;         // N_LOANS
  float* agg_lc    = rs_in_ll  + N_LOANS;         // N_CLIENTS*FD
  float* agg_cl    = agg_lc + (size_t)N_CLIENTS * FD;   // N_LOANS*FD
  float* agg_ll    = agg_cl + (size_t)N_LOANS * FD;     // N_LOANS*FD
  float* h_clients = agg_ll + (size_t)N_LOANS * FD;     // N_CLIENTS*FD (no zero needed)

  const size_t n_deg  = (size_t)3 * N_LOANS + 2 * N_CLIENTS + N_LOANS; // 1.5M
  const size_t n_zero = n_deg + (size_t)N_CLIENTS * FD + (size_t)2 * N_LOANS * FD;

  float* out_loans   = (float*)d_out;
  float* out_clients = out_loans + (size_t)N_LOANS * FD;

  hipMemsetAsync(d_ws, 0, n_zero * sizeof(float), stream);

  // degrees
  count_deg_kernel<<<(E_LC + 255) / 256, 256, 0, stream>>>(src_lc, E_LC, rs_out_lc);
  count_deg_kernel<<<(E_LC + 255) / 256, 256, 0, stream>>>(dst_lc, E_LC, rs_in_lc);
  count_deg_kernel<<<(E_CL + 255) / 256, 256, 0, stream>>>(src_cl, E_CL, rs_out_cl);
  count_deg_kernel<<<(E_CL + 255) / 256, 256, 0, stream>>>(dst_cl, E_CL, rs_in_cl);
  count_deg_kernel<<<(E_LL + 255) / 256, 256, 0, stream>>>(src_ll, E_LL, rs_out_ll);
  count_deg_kernel<<<(E_LL + 255) / 256, 256, 0, stream>>>(dst_ll, E_LL, rs_in_ll);
  rsqrt_clip_kernel<<<((int)n_deg + 255) / 256, 256, 0, stream>>>(rs_out_lc, (int)n_deg);

  // shared lc aggregation (feeds both W1_lc and W2_lc GEMMs)
  spmm_kernel<<<(E_LC + 7) / 8, 256, 0, stream>>>(x_loans, rs_out_lc, src_lc, dst_lc, E_LC, agg_lc);
  // loan->loan aggregation (independent)
  spmm_kernel<<<(E_LL + 7) / 8, 256, 0, stream>>>(x_loans, rs_out_ll, src_ll, dst_ll, E_LL, agg_ll);

  // clients GEMM pass: h_clients (relu) + out_clients
  gemm_clients_kernel<<<N_CLIENTS / 16, 256, 0, stream>>>(
      agg_lc, rs_in_lc, W1_lc, b1_lc, W2_lc, b2_lc, h_clients, out_clients);

  // client->loan aggregation of h_clients
  spmm_kernel<<<(E_CL + 7) / 8, 256, 0, stream>>>(h_clients, rs_out_cl, src_cl, dst_cl, E_CL, agg_cl);

  // loans GEMM pass: out_loans = cl-part + ll-part
  gemm_loans_kernel<<<N_LOANS / 16, 256, 0, stream>>>(
      agg_cl, rs_in_cl, W2_cl, agg_ll, rs_in_ll, W2_ll, b2_cl, b2_ll, out_loans);
}